// LabelSmoothingLoss_5248450036239
// MI455X (gfx1250) — compile-verified
//
#include <hip/hip_runtime.h>

typedef __attribute__((ext_vector_type(2))) float v2f;
typedef __attribute__((ext_vector_type(8))) float v8f;

#define N_CLASSES 101
#define K_PAD     104   // next multiple of 4 (WMMA K=4 steps), zero padded

// ---------------------------------------------------------------------------
// Main kernel: one wave handles 16 rows at a time.
//   loss contribution per 16-row tile = Sum_m [ S1(t_m) - dot(p_{t_m}, out_m) ]
// dot() via V_WMMA_F32_16X16X4_F32, diagonal of 16x16 D extracted per lane.
// ---------------------------------------------------------------------------
__global__ void __launch_bounds__(256)
lsl_main_kernel(const float* __restrict__ out,
                const long long* __restrict__ tgt,
                float* __restrict__ partial,
                int B)
{
    __shared__ float pTab[N_CLASSES][K_PAD];   // 42016 B: p[t][c], zero padded
    __shared__ float s1Tab[N_CLASSES];         // S1(t) = sum_c p*logp
    __shared__ float wsum[8];

    const int tid = threadIdx.x;

    // ---- build per-target tables (startup only; max logit is 0 so exp stable)
    const float inv2s2 = 1.0f / (2.0f * 5.05f * 5.05f);  // STD = 0.05*101
    for (int t = tid; t < N_CLASSES; t += blockDim.x) {
        float Z = 0.0f;
        for (int c = 0; c < N_CLASSES; ++c) {
            float d = (float)(c - t);
            Z += expf(-d * d * inv2s2);
        }
        float logZ = logf(Z);
        float invZ = 1.0f / Z;
        float s1 = 0.0f;
        for (int c = 0; c < K_PAD; ++c) {
            float p = 0.0f;
            if (c < N_CLASSES) {
                float d = (float)(c - t);
                float l = -d * d * inv2s2;
                p = expf(l) * invZ;
                s1 += p * (l - logZ);
            }
            pTab[t][c] = p;
        }
        s1Tab[t] = s1;
    }
    __syncthreads();

    const int lane       = tid & 31;
    const int waveInBlk  = tid >> 5;
    const int wavesPerBlk = blockDim.x >> 5;
    const int gwave  = blockIdx.x * wavesPerBlk + waveInBlk;
    const int nwaves = gridDim.x * wavesPerBlk;
    const int ntiles = B >> 4;              // B = 262144 -> 16384 tiles of 16 rows

    // ISA 32-bit A 16x4 layout: lane L holds row M = L%16;
    // VGPR0/1 hold K = {0,1} (lanes 0-15) or K = {2,3} (lanes 16-31).
    const int m     = lane & 15;
    const int khalf = (lane >> 4) << 1;     // 0 or 2

    float acc = 0.0f;

    for (int tile = gwave; tile < ntiles; tile += nwaves) {
        const int row = (tile << 4) + m;
        const int t   = (int)tgt[row];      // 0..100 (int64 input)
        const float* rowp = out + (long long)row * N_CLASSES;
        const float* pt   = &pTab[t][0];

        v8f c = {};
        #pragma unroll 4
        for (int kb = 0; kb < K_PAD; kb += 4) {
            const int k0 = kb + khalf;
            v2f a, b;
            a.x = (k0     < N_CLASSES) ? rowp[k0]     : 0.0f;
            a.y = (k0 + 1 < N_CLASSES) ? rowp[k0 + 1] : 0.0f;
            b.x = pt[k0];                   // table zero-padded: no guard
            b.y = pt[k0 + 1];
            // D = A(16x4 f32) x B(4x16 f32) + C  -> v_wmma_f32_16x16x4_f32
            c = __builtin_amdgcn_wmma_f32_16x16x4_f32(
                    false, a, false, b, (short)0, c, false, false);
        }

        // Diagonal D[m][m]: m<8 -> lane m, VGPR m ; m>=8 -> lane m+16, VGPR m-8.
        float diag = 0.0f;
        #pragma unroll
        for (int v = 0; v < 8; ++v) {
            bool take = (lane == v) || (lane == v + 24);
            diag += take ? c[v] : 0.0f;
        }
        const float s1part = (lane < 16) ? s1Tab[t] : 0.0f;  // one add per row
        acc += s1part - diag;
    }

    // wave32 reduction
    #pragma unroll
    for (int off = 16; off > 0; off >>= 1)
        acc += __shfl_xor(acc, off, 32);
    if (lane == 0) wsum[waveInBlk] = acc;
    __syncthreads();

    if (tid == 0) {
        float s = 0.0f;
        for (int w = 0; w < wavesPerBlk; ++w) s += wsum[w];
        partial[blockIdx.x] = s;
    }
}

// ---------------------------------------------------------------------------
// Deterministic final reduction (single block) -> d_out[0]
// ---------------------------------------------------------------------------
__global__ void __launch_bounds__(256)
lsl_reduce_kernel(const float* __restrict__ partial, int n, float* __restrict__ out)
{
    __shared__ float sh[256];
    float s = 0.0f;
    for (int i = threadIdx.x; i < n; i += 256) s += partial[i];
    sh[threadIdx.x] = s;
    __syncthreads();
    #pragma unroll
    for (int stride = 128; stride > 0; stride >>= 1) {
        if (threadIdx.x < stride) sh[threadIdx.x] += sh[threadIdx.x + stride];
        __syncthreads();
    }
    if (threadIdx.x == 0) out[0] = sh[0];
}

extern "C" void kernel_launch(void* const* d_in, const int* in_sizes, int n_in,
                              void* d_out, int out_size, void* d_ws, size_t ws_size,
                              hipStream_t stream)
{
    const float*     output = (const float*)d_in[0];
    const long long* target = (const long long*)d_in[1];
    const int B = in_sizes[1];                 // 262144

    float* partial = (float*)d_ws;
    int blocks = 1024;                         // 8192 waves, 2 tiles each
    if (ws_size < (size_t)blocks * sizeof(float))
        blocks = (int)(ws_size / sizeof(float));
    if (blocks < 1) blocks = 1;

    lsl_main_kernel<<<blocks, 256, 0, stream>>>(output, target, partial, B);
    lsl_reduce_kernel<<<1, 256, 0, stream>>>(partial, blocks, (float*)d_out);
}